// EdgeWeightLearner_31842887533249
// MI455X (gfx1250) — compile-verified
//
#include <hip/hip_runtime.h>
#include <hip/hip_bf16.h>

typedef float v2f __attribute__((ext_vector_type(2)));
typedef float v8f __attribute__((ext_vector_type(8)));

#define D_DIM 128

// ---------------------------------------------------------------------------
// Kernel 1: per-node projection  s[n] = x[n]·W[0:128], t[n] = x[n]·W[128:256]
// One wave32 handles a 16-row tile of x via V_WMMA_F32_16X16X4_F32,
// accumulating over 32 chunks of K=4. B has W1 in column 0, W2 in column 1.
// Branch-free inner loop: all lanes load a valid (clamped) W address, unused
// B columns are zeroed by a precomputed multiplicative mask, so EXEC stays
// all-1s with no saveexec traffic around the WMMA.
// ---------------------------------------------------------------------------
__global__ void node_proj_wmma(const float* __restrict__ x,
                               const float* __restrict__ W,
                               float* __restrict__ s,
                               float* __restrict__ t,
                               int N) {
    const int lane   = threadIdx.x & 31;
    const int waveId = (int)(blockIdx.x * (blockDim.x >> 5) + (threadIdx.x >> 5));
    const int rowBase = waveId * 16;
    if (rowBase >= N) return;                 // wave-uniform exit

    const int col   = lane & 15;              // N index (0..15) for B/C/D
    const int khalf = (lane >> 4) * 2;        // K sub-offset: 0 (lo half) / 2 (hi half)

    // Clamp row for loads so EXEC stays all-1s through the WMMA loop;
    // out-of-range results are masked at the store.
    int mrow = rowBase + col;
    if (mrow >= N) mrow = N - 1;
    const float* xrow = x + (size_t)mrow * D_DIM + khalf;

    // Every lane gets a valid B address (lanes >=2 alias lane 0/1's data and
    // are zeroed by bmask), so the B load needs no EXEC predication.
    const float* brow  = W + (col & 1) * D_DIM + khalf;
    const float  bmask = (col < 2) ? 1.0f : 0.0f;

    v8f c = {};
    #pragma unroll 8
    for (int k = 0; k < D_DIM; k += 4) {
        // A-matrix 16x4 f32: lane holds x[mrow, k+khalf .. k+khalf+1] (8B aligned)
        v2f a = *(const v2f*)(xrow + k);

        // B-matrix 4x16 f32: column 0 = W1 chunk, column 1 = W2 chunk, rest 0
        v2f b = *(const v2f*)(brow + k);
        b.x *= bmask;
        b.y *= bmask;

        // D = A x B + C   (8 args: neg_a, A, neg_b, B, c_mod, C, reuse_a, reuse_b)
        c = __builtin_amdgcn_wmma_f32_16x16x4_f32(
                false, a, false, b, (short)0, c, false, false);
    }

    // D layout (32-bit 16x16): lanes 0-15 -> M = vgpr (0..7), lanes 16-31 -> M = vgpr+8.
    // Column 0 (s) lives in lanes 0/16; column 1 (t) lives in lanes 1/17.
    if (col < 2) {
        float* dst = (col == 0) ? s : t;
        const int mOff = (lane >> 4) * 8;     // 0 for lanes 0/1, 8 for lanes 16/17
        #pragma unroll
        for (int i = 0; i < 8; ++i) {
            int r = rowBase + mOff + i;
            if (r < N) dst[r] = c[i];
        }
    }
}

// ---------------------------------------------------------------------------
// Kernel 2: per-edge logits + sigmoid.  wght[e] = sigmoid(s[row[e]] + t[col[e]])
// s/t are 200 KB each -> resident in L2 (192 MB) -> random gathers are cheap.
// ---------------------------------------------------------------------------
__global__ void edge_sigmoid(const int* __restrict__ row,
                             const int* __restrict__ colv,
                             const float* __restrict__ s,
                             const float* __restrict__ t,
                             float* __restrict__ wght,
                             int E) {
    int e = (int)(blockIdx.x * blockDim.x + threadIdx.x);
    if (e >= E) return;
    float logit = s[row[e]] + t[colv[e]];
    wght[e] = 1.0f / (1.0f + __expf(-logit));
}

// ---------------------------------------------------------------------------
// Kernel 3: out[e] = wght[e] * wght[full_right_idx[e]]
// ---------------------------------------------------------------------------
__global__ void edge_pair(const float* __restrict__ wght,
                          const int* __restrict__ fri,
                          float* __restrict__ out,
                          int E) {
    int e = (int)(blockIdx.x * blockDim.x + threadIdx.x);
    if (e >= E) return;
    out[e] = wght[e] * wght[fri[e]];
}

extern "C" void kernel_launch(void* const* d_in, const int* in_sizes, int n_in,
                              void* d_out, int out_size, void* d_ws, size_t ws_size,
                              hipStream_t stream) {
    const float* x   = (const float*)d_in[0];   // [N, 128]
    const float* W   = (const float*)d_in[1];   // [1, 256]
    const int*   ei  = (const int*)  d_in[2];   // [2, E] flattened: row then col
    const int*   fri = (const int*)  d_in[3];   // [E]

    float* out = (float*)d_out;                 // [E]

    const int N = in_sizes[0] / D_DIM;
    const int E = in_sizes[3];

    // Workspace layout: s[N] | t[N] | wght[E]   (all f32)
    float* s    = (float*)d_ws;
    float* t    = s + N;
    float* wght = t + N;

    // Projection: one wave per 16 rows, 8 waves (256 threads) per block.
    const int waves      = (N + 15) / 16;
    const int projBlocks = (waves + 7) / 8;
    node_proj_wmma<<<projBlocks, 256, 0, stream>>>(x, W, s, t, N);

    const int edgeBlocks = (E + 255) / 256;
    edge_sigmoid<<<edgeBlocks, 256, 0, stream>>>(ei, ei + E, s, t, wght, E);
    edge_pair  <<<edgeBlocks, 256, 0, stream>>>(wght, fri, out, E);
}